// LSTMCell_27118423507724
// MI455X (gfx1250) — compile-verified
//
#include <hip/hip_runtime.h>

// Problem constants (from reference): N=8192, n_in=256, m=256, k=8
#define NROWS 8192
#define NIN   256
#define MDIM  256
#define KCAT  8
#define KDIM  512          // GEMM reduction dim = n_in + m
#define WCOLS 8192         // 4*m*k
#define GATE_STRIDE 2048   // m*k

typedef __attribute__((ext_vector_type(2))) float v2f;
typedef __attribute__((ext_vector_type(8))) float v8f;

__device__ __forceinline__ float sigf(float x) { return 1.0f / (1.0f + __expf(-x)); }

// ---------------------------------------------------------------------------
// Kernel 1: logit_z = x_t@W_xz + b_xz + h_0@W_hz ; q_z = softmax(logit)
//           z = softmax((logit + gumbel(u)) / tau).  Tiny (67 MFLOP) — VALU.
// One block = 32 rows; thread t -> (row t/8, category t%8).
// ---------------------------------------------------------------------------
__global__ __launch_bounds__(256)
void logits_softmax_kernel(const float* __restrict__ x_t,
                           const float* __restrict__ h_0,
                           const float* __restrict__ u,
                           const float* __restrict__ W_xz,
                           const float* __restrict__ W_hz,
                           const float* __restrict__ b_xz,
                           const int*   __restrict__ tau_p,
                           float* __restrict__ z_out,
                           float* __restrict__ q_out)
{
    __shared__ float L[32][KCAT];
    const int t   = threadIdx.x;
    const int row = blockIdx.x * 32 + (t >> 3);
    const int c   = t & 7;

    float acc = b_xz[c];
    const float* xr = x_t + (size_t)row * NIN;
    const float* hr = h_0 + (size_t)row * NIN;
#pragma unroll 4
    for (int i = 0; i < NIN; ++i) acc += xr[i] * W_xz[i * KCAT + c];
#pragma unroll 4
    for (int i = 0; i < NIN; ++i) acc += hr[i] * W_hz[i * KCAT + c];
    L[t >> 3][c] = acc;
    __syncthreads();

    const float tau = (float)(*tau_p);
    const float* ur = u + (size_t)row * KCAT;

    float lg[KCAT], y[KCAT];
    float mq = -1e30f, mz = -1e30f;
#pragma unroll
    for (int j = 0; j < KCAT; ++j) {
        float lj = L[t >> 3][j];
        lg[j] = lj;
        float g = -__logf(-__logf(ur[j] + 1e-10f) + 1e-10f);
        y[j] = (lj + g) / tau;
        mq = fmaxf(mq, lj);
        mz = fmaxf(mz, y[j]);
    }
    float sq = 0.0f, sz = 0.0f;
#pragma unroll
    for (int j = 0; j < KCAT; ++j) {
        sq += __expf(lg[j] - mq);
        sz += __expf(y[j]  - mz);
    }
    q_out[row * KCAT + c] = __expf(lg[c] - mq) / sq;
    z_out[row * KCAT + c] = __expf(y[c]  - mz) / sz;
}

// ---------------------------------------------------------------------------
// Kernel 2: fused big GEMM + LSTM epilogue + k-category reduction.
//   gates[n, g*2048 + kk*256 + mm] = [x|h] @ [Wx4;Wh4] + b4   (never stored)
//   C = sig(F)*c0 + sig(I)*tanh(G); H = sig(O)*tanh(C)
//   h_t[n,mm] = sum_kk z[n,kk]*H ; c_t likewise.
// Block = 32 rows x 16 mm, 8 waves, wave w owns category kk=w.
// fp32 WMMA 16x16x4, K-loop split into the Wx half and the Wh half so each
// inner loop has a loop-invariant W base pointer (no branch / cselect in the
// hot loop).  8 wmma per k-step per wave.
// ---------------------------------------------------------------------------
__global__ __launch_bounds__(256)
void fused_gates_kernel(const float* __restrict__ x_t,
                        const float* __restrict__ h_0,
                        const float* __restrict__ c_0,
                        const float* __restrict__ Wx,   // [256, 8192]
                        const float* __restrict__ Wh,   // [256, 8192]
                        const float* __restrict__ b4,   // [8192]
                        const float* __restrict__ z,    // [N, 8]
                        float* __restrict__ h_t,
                        float* __restrict__ c_t)
{
    __shared__ float As[32][KDIM + 2];   // +2 pad: kills stride-512 bank conflicts
    __shared__ float redH[32][16];
    __shared__ float redC[32][16];

    const int t       = threadIdx.x;
    const int row_blk = blockIdx.x >> 4;   // 0..255
    const int mm_blk  = blockIdx.x & 15;   // 0..15
    const int n0      = row_blk * 32;
    const int mm0     = mm_blk * 16;

    // zero the z-weighted reduction buffers
#pragma unroll
    for (int i = 0; i < 2; ++i) {
        ((float*)redH)[t + i * 256] = 0.0f;
        ((float*)redC)[t + i * 256] = 0.0f;
    }

    // Stage A tile: rows n0..n0+31, K = [x_t row | h_0 row] (coalesced)
    for (int idx = t; idx < 32 * KDIM; idx += 256) {
        int r = idx >> 9;          // / 512
        int c = idx & (KDIM - 1);  // % 512
        As[r][c] = (c < NIN) ? x_t[(size_t)(n0 + r) * NIN + c]
                             : h_0[(size_t)(n0 + r) * NIN + (c - NIN)];
    }
    __syncthreads();

    const int lane = t & 31;
    const int wave = t >> 5;             // category kk = wave (0..7)
    const int ln16 = lane & 15;
    const int koff = (lane >> 4) << 1;   // lanes 0-15 -> K {0,1}; 16-31 -> {2,3}

    // per-gate B column for this lane; bias broadcast into accumulators
    int col[4];
    v8f acc[2][4];
#pragma unroll
    for (int g = 0; g < 4; ++g) {
        col[g] = g * GATE_STRIDE + wave * MDIM + mm0 + ln16;
        float b = b4[col[g]];
#pragma unroll
        for (int s = 0; s < 2; ++s)
#pragma unroll
            for (int j = 0; j < 8; ++j) acc[s][g][j] = b;
    }

    // Two K-halves: kbase=0 streams Wx, kbase=256 streams Wh.  Each half has
    // a loop-invariant base pointer -> no branch/cselect in the inner loop.
#pragma unroll 1
    for (int half = 0; half < 2; ++half) {
        const float* __restrict__ Wbase = (half == 0) ? Wx : Wh;
        const int kbase = half * NIN;
#pragma unroll 1
        for (int kr = 0; kr < NIN; kr += 4) {
            const int k0 = kbase + kr;
            const float* Wb = Wbase + (size_t)kr * WCOLS;

            // A fragments (16x4 fp32 layout) from LDS, shared by all 8 waves
            v2f a[2];
#pragma unroll
            for (int s = 0; s < 2; ++s) {
                const float* ap = &As[s * 16 + ln16][k0 + koff];
                a[s].x = ap[0];
                a[s].y = ap[1];
            }
            // B fragments (4x16 fp32 layout) straight from L2-resident W
            v2f b[4];
#pragma unroll
            for (int g = 0; g < 4; ++g) {
                const float* bp = Wb + (size_t)koff * WCOLS + col[g];
                b[g].x = bp[0];
                b[g].y = bp[WCOLS];
            }
            __builtin_prefetch(Wb + (size_t)4 * WCOLS + col[0], 0, 1);

#pragma unroll
            for (int s = 0; s < 2; ++s)
#pragma unroll
                for (int g = 0; g < 4; ++g)
                    acc[s][g] = __builtin_amdgcn_wmma_f32_16x16x4_f32(
                        false, a[s], false, b[g], (short)0, acc[s][g], false, false);
        }
    }

    // Epilogue: C/D layout => VGPR j holds M = j (lanes 0-15) or j+8 (lanes 16-31)
#pragma unroll
    for (int s = 0; s < 2; ++s) {
#pragma unroll
        for (int j = 0; j < 8; ++j) {
            int rloc = s * 16 + j + ((lane >> 4) << 3);
            int n    = n0 + rloc;
            float I = acc[s][0][j];
            float G = acc[s][1][j];
            float F = acc[s][2][j];
            float O = acc[s][3][j];
            float c0v = c_0[(size_t)n * MDIM + mm0 + ln16];
            float Ct  = sigf(F) * c0v + sigf(I) * tanhf(G);
            float Ht  = sigf(O) * tanhf(Ct);
            float zv  = z[(size_t)n * KCAT + wave];
            atomicAdd(&redH[rloc][ln16], zv * Ht);   // ds_add_f32
            atomicAdd(&redC[rloc][ln16], zv * Ct);
        }
    }
    __syncthreads();

    // coalesced store of the 32x16 h_t / c_t tiles
#pragma unroll
    for (int i = 0; i < 2; ++i) {
        int idx = t + i * 256;    // 0..511
        int r   = idx >> 4;
        int c   = idx & 15;
        h_t[(size_t)(n0 + r) * MDIM + mm0 + c] = redH[r][c];
        c_t[(size_t)(n0 + r) * MDIM + mm0 + c] = redC[r][c];
    }
}

// ---------------------------------------------------------------------------
extern "C" void kernel_launch(void* const* d_in, const int* in_sizes, int n_in_args,
                              void* d_out, int out_size, void* d_ws, size_t ws_size,
                              hipStream_t stream)
{
    const float* x_t  = (const float*)d_in[0];
    const float* h_0  = (const float*)d_in[1];
    const float* c_0  = (const float*)d_in[2];
    const float* u    = (const float*)d_in[3];
    const float* W_xz = (const float*)d_in[4];
    const float* W_hz = (const float*)d_in[5];
    const float* Wx4  = (const float*)d_in[6];
    const float* Wh4  = (const float*)d_in[7];
    const float* b_xz = (const float*)d_in[8];
    const float* b4   = (const float*)d_in[9];
    const int*   tau  = (const int*)d_in[10];

    // outputs concatenated flat in return order: z, q_z, h_t, c_t
    float* z_out = (float*)d_out;
    float* q_out = z_out + (size_t)NROWS * KCAT;
    float* h_t   = q_out + (size_t)NROWS * KCAT;
    float* c_t   = h_t   + (size_t)NROWS * MDIM;

    logits_softmax_kernel<<<NROWS / 32, 256, 0, stream>>>(
        x_t, h_0, u, W_xz, W_hz, b_xz, tau, z_out, q_out);

    fused_gates_kernel<<<(NROWS / 32) * (MDIM / 16), 256, 0, stream>>>(
        x_t, h_0, c_0, Wx4, Wh4, b4, z_out, h_t, c_t);
}